// TrellisQuantizer_9637906612612
// MI455X (gfx1250) — compile-verified
//
#include <hip/hip_runtime.h>

// Trellis dequant: out[i] = tlut[(h>>5)&1023] * signflips(h), h = x*(x+1), x = encoded[i].
// Pure bandwidth problem (~403 MB streamed => ~17us at 23.3 TB/s). Codebook (8 KB)
// staged into LDS via CDNA5 async global->LDS copies; all global traffic 16B coalesced.

#define TLUT_ENTRIES 1024
#define LDS_BYTES (TLUT_ENTRIES * 8)   // 1024 x float2 = 8 KB
#define BLOCK 256

// The async builtin takes int4* in global (AS1) / LDS (AS3) address spaces.
typedef int v4i __attribute__((ext_vector_type(4)));
typedef __attribute__((address_space(1))) v4i* gv4_t;
typedef __attribute__((address_space(3))) v4i* lv4_t;

__device__ __forceinline__ float2 dequant_one(unsigned x, const float2* __restrict__ lut2) {
    unsigned h = x * (x + 1u);                 // hash, wraps mod 2^32 like the uint32 reference
    float2 v = lut2[(h >> 5) & 1023u];         // sel = (h >> (16-10-1)) & (2^10 - 1)
    unsigned s0 = (h & 0x8000u) << 16;         // bit 15 -> sign bit
    unsigned s1 = (h & 0x2000u) << 18;         // bit 13 -> sign bit
    float2 r;
    r.x = __uint_as_float(__float_as_uint(v.x) ^ s0);
    r.y = __uint_as_float(__float_as_uint(v.y) ^ s1);
    return r;
}

__global__ __launch_bounds__(BLOCK) void trellis_dequant_kernel(
    const float* __restrict__ tlut,   // [1024, 2] f32
    const int*   __restrict__ enc,    // [N] int32 codes in [0, 65536)
    float*       __restrict__ out,    // [N, 2] f32
    int n4)                           // N / 4
{
    __shared__ __align__(16) float lut[TLUT_ENTRIES * 2];

    // ---- Stage the 8 KB codebook into LDS (CDNA5 async global->LDS path) ----
#if __has_builtin(__builtin_amdgcn_global_load_async_to_lds_b128)
    {
        gv4_t g = (gv4_t)tlut;   // addrspacecast generic -> global, const dropped
        lv4_t l = (lv4_t)lut;    // addrspacecast generic -> LDS
        // 512 x 16B chunks, per-lane addresses; 2 async instructions per wave.
        #pragma unroll
        for (int i = threadIdx.x; i < (LDS_BYTES / 16); i += BLOCK) {
            __builtin_amdgcn_global_load_async_to_lds_b128(g + i, l + i, 0, 0);
        }
#if __has_builtin(__builtin_amdgcn_s_wait_asynccnt)
        __builtin_amdgcn_s_wait_asynccnt(0);
#else
        asm volatile("s_wait_asynccnt 0" ::: "memory");
#endif
    }
#else
    // Fallback: plain coalesced fill (b32 loads + ds stores).
    for (int i = threadIdx.x; i < TLUT_ENTRIES * 2; i += BLOCK)
        lut[i] = tlut[i];
#endif
    __syncthreads();

    const float2* __restrict__ lut2 = (const float2*)lut;

    int t = blockIdx.x * BLOCK + threadIdx.x;
    if (t >= n4) return;

    // 4 codes per thread: one 16B load, two 16B stores, all coalesced.
    int4 e = ((const int4*)enc)[t];

    float2 r0 = dequant_one((unsigned)e.x, lut2);
    float2 r1 = dequant_one((unsigned)e.y, lut2);
    float2 r2 = dequant_one((unsigned)e.z, lut2);
    float2 r3 = dequant_one((unsigned)e.w, lut2);

    float4* __restrict__ out4 = (float4*)out;
    out4[2 * t + 0] = make_float4(r0.x, r0.y, r1.x, r1.y);
    out4[2 * t + 1] = make_float4(r2.x, r2.y, r3.x, r3.y);
}

extern "C" void kernel_launch(void* const* d_in, const int* in_sizes, int n_in,
                              void* d_out, int out_size, void* d_ws, size_t ws_size,
                              hipStream_t stream) {
    const float* tlut = (const float*)d_in[0];   // [1024, 2] f32
    const int*   enc  = (const int*)d_in[1];     // [16384 * 2048] int32
    float*       out  = (float*)d_out;           // [16384 * 2048 * 2] f32

    int n  = in_sizes[1];      // 33,554,432 codes
    int n4 = n >> 2;           // 4 codes per thread (n is a multiple of 4: 2048 cols)

    int blocks = (n4 + BLOCK - 1) / BLOCK;
    trellis_dequant_kernel<<<blocks, BLOCK, 0, stream>>>(tlut, enc, out, n4);
}